// QuantumLayer_37074157699082
// MI455X (gfx1250) — compile-verified
//
#include <hip/hip_runtime.h>

// Fused "quantum layer": out = (prod-of-cos circuit)(x @ w_in^T + b_in) @ w_out^T + b_out
// The 16-qubit statevector sim collapses to ev[b,i] = prod_{k in S_i} cos(h[b,k]);
// q_weights (RZ phases) provably cancel in the PauliZ expectations.

typedef __attribute__((ext_vector_type(2))) float v2f;
typedef __attribute__((ext_vector_type(8))) float v8f;

__global__ __launch_bounds__(32) void qlayer_fused(
    const float* __restrict__ x,      // [512,128]
    const float* __restrict__ w_in,   // [16,128]
    const float* __restrict__ b_in,   // [16]
    const float* __restrict__ w_out,  // [128,16]
    const float* __restrict__ b_out,  // [128]
    float* __restrict__ out)          // [512,128]
{
    __shared__ float ch[16][20];      // [batch row m][qubit k], row-padded (16B aligned, bank-clean)

    const int lane = threadIdx.x;     // wave32
    const int half = lane >> 4;       // 0 => K/M low half, 1 => high half (WMMA f32 layouts)
    const int l    = lane & 15;
    const int r0   = blockIdx.x << 4; // 16 batch rows per workgroup (512/16 = 32 blocks)

    // ---------------- GEMM1: h[16,16] = x_tile[16,128] @ w_in^T, K=128 as 32 chained WMMAs
    // A 16x4 f32 layout: lane L: VGPR0=A[L,4k+0],VGPR1=A[L,4k+1]; lane L+16: K=4k+{2,3}
    // B  4x16 f32 layout mirrors A transposed: B[k][n] = w_in[n][4k0+k]
    v8f acc = {0.f, 0.f, 0.f, 0.f, 0.f, 0.f, 0.f, 0.f};
    const float* ax = x    + (size_t)(r0 + l) * 128 + 2 * half;
    const float* bw = w_in + (size_t)l        * 128 + 2 * half;
    #pragma unroll
    for (int k = 0; k < 32; ++k) {
        v2f a = *(const v2f*)(ax + 4 * k);
        v2f b = *(const v2f*)(bw + 4 * k);
        acc = __builtin_amdgcn_wmma_f32_16x16x4_f32(false, a, false, b,
                                                    (short)0, acc, false, false);
    }

    // bias + cos -> LDS. C/D layout: lane has N=l; VGPR j holds row M = j + 8*half.
    const float bi = b_in[l];
    #pragma unroll
    for (int j = 0; j < 8; ++j)
        ch[j + 8 * half][l] = __cosf(acc[j] + bi);
    __syncthreads();

    // ---------------- circuit closed form: ev[m][i] = prod_{k in S_i} cos(h[m][k])
    // S_i = { k <= i : (i-k) mod 4 in {0,1} }  (CNOT chain ^3 over GF(2))
    float chr[16];
    #pragma unroll
    for (int q = 0; q < 16; ++q) chr[q] = ch[l][q];   // lane owns row m=l (broadcast-safe)

    // Build ev directly in GEMM2 A-operand layout: lane l holds ev[l, 4k+2*half+{0,1}]
    v2f evA[4];
    #pragma unroll
    for (int k = 0; k < 4; ++k) {
        #pragma unroll
        for (int t = 0; t < 2; ++t) {
            const int i = 4 * k + 2 * half + t;
            const unsigned mask =
                ((0x1111u << (i & 3)) | (0x1111u << ((i + 3) & 3))) & ((2u << i) - 1u);
            float p = 1.0f;
            #pragma unroll
            for (int q = 0; q < 16; ++q)
                p *= ((mask >> q) & 1u) ? chr[q] : 1.0f;
            evA[k][t] = p;
        }
    }

    // ---------------- GEMM2: out_tile[16,128] = ev[16,16] @ w_out^T + b_out
    // 8 N-tiles of 16 columns, K=16 as 4 chained WMMAs. B[k][n] = w_out[16t+n][4k0+k].
    #pragma unroll
    for (int t = 0; t < 8; ++t) {
        v8f acc2 = {0.f, 0.f, 0.f, 0.f, 0.f, 0.f, 0.f, 0.f};
        #pragma unroll
        for (int k = 0; k < 4; ++k) {
            v2f b2 = *(const v2f*)(w_out + (size_t)(16 * t + l) * 16 + 4 * k + 2 * half);
            acc2 = __builtin_amdgcn_wmma_f32_16x16x4_f32(false, evA[k], false, b2,
                                                         (short)0, acc2, false, false);
        }
        const float bo = b_out[16 * t + l];
        #pragma unroll
        for (int j = 0; j < 8; ++j)
            out[(size_t)(r0 + j + 8 * half) * 128 + 16 * t + l] = acc2[j] + bo;
    }
}

extern "C" void kernel_launch(void* const* d_in, const int* in_sizes, int n_in,
                              void* d_out, int out_size, void* d_ws, size_t ws_size,
                              hipStream_t stream) {
    (void)in_sizes; (void)n_in; (void)d_ws; (void)ws_size; (void)out_size;
    const float* x     = (const float*)d_in[0];  // [512,128]
    const float* w_in  = (const float*)d_in[1];  // [16,128]
    const float* b_in  = (const float*)d_in[2];  // [16]
    // d_in[3] = q_weights [3,16]: provably does not affect the output (diagonal phases)
    const float* w_out = (const float*)d_in[4];  // [128,16]
    const float* b_out = (const float*)d_in[5];  // [128]
    float* out = (float*)d_out;                  // [512,128]

    qlayer_fused<<<dim3(32), dim3(32), 0, stream>>>(x, w_in, b_in, w_out, b_out, out);
}